// net_LSTM_21534966022336
// MI455X (gfx1250) — compile-verified
//
#include <hip/hip_runtime.h>
#include <stdint.h>

// Scalar LSTM scan (I=H=1): one thread per batch element, serial over T.
// Chain-optimized with v_tanh_f32; x staged through LDS with CDNA5 async
// global->LDS copies, double-buffered per 32-timestep chunk.

#define BLK   256
#define CHUNK 32   // timesteps staged per LDS buffer (32 async b32 per thread, ASYNCcnt<=32)

typedef __attribute__((address_space(1))) int  as1_int;
typedef __attribute__((address_space(3))) int  as3_int;

// ---------------- fast tanh: hardware v_tanh_f32 on gfx1250 ----------------
__device__ __forceinline__ float fast_tanh(float z) {
#if __has_builtin(__builtin_amdgcn_tanhf)
    return __builtin_amdgcn_tanhf(z);
#else
    // tanh(z) = 1 - 2/(exp2(2*log2e*z) + 1); saturates correctly at +/-inf
    float p = __builtin_amdgcn_exp2f(z * 2.8853900817779268f);
    return 1.0f - 2.0f * __builtin_amdgcn_rcpf(p + 1.0f);
#endif
}

// ------------- CDNA5 async global->LDS copy (ASYNCcnt tracked) -------------
__device__ __forceinline__ void async_ld_f32(const float* gsrc, uint32_t lds_off) {
#if __has_builtin(__builtin_amdgcn_global_load_async_to_lds_b32)
    __builtin_amdgcn_global_load_async_to_lds_b32(
        (as1_int*)(uintptr_t)gsrc,
        (as3_int*)lds_off,
        0 /*offset*/, 0 /*cpol*/);
#else
    asm volatile("global_load_async_to_lds_b32 %0, %1, off"
                 :: "v"(lds_off), "v"(gsrc) : "memory");
#endif
}

__device__ __forceinline__ void wait_async0() {
#if __has_builtin(__builtin_amdgcn_s_wait_asynccnt)
    __builtin_amdgcn_s_wait_asynccnt(0);
#else
    asm volatile("s_wait_asynccnt 0x0" ::: "memory");
#endif
}

// ---------------------------------------------------------------------------
__global__ __launch_bounds__(BLK) void lstm_scan_kernel(
    const float* __restrict__ x,     // [T, B]
    const float* __restrict__ h0,    // [B]
    const float* __restrict__ c0,    // [B]
    const float* __restrict__ w_ih,  // [4] gate order i,f,g,o
    const float* __restrict__ w_hh,  // [4]
    float* __restrict__ out,         // [T, B]
    float* __restrict__ hN,          // [B]
    float* __restrict__ cN,          // [B]
    int T, int B)
{
    __shared__ float sbuf[2][CHUNK * BLK];

    const int tid    = threadIdx.x;
    const int b      = blockIdx.x * BLK + tid;
    const bool valid = (b < B);

    // sigmoid(z) = 0.5*tanh(z/2)+0.5  -> fold the 1/2 into the weights
    const float si_x = 0.5f * w_ih[0], sf_x = 0.5f * w_ih[1];
    const float g_x  =        w_ih[2], so_x = 0.5f * w_ih[3];
    const float si_h = 0.5f * w_hh[0], sf_h = 0.5f * w_hh[1];
    const float g_h  =        w_hh[2], so_h = 0.5f * w_hh[3];

    float h = valid ? h0[b] : 0.0f;
    float c = valid ? c0[b] : 0.0f;

    const uint32_t lds0 = (uint32_t)(uintptr_t)&sbuf[0][0];
    const uint32_t lds1 = (uint32_t)(uintptr_t)&sbuf[1][0];

    const int nfull = T / CHUNK;           // full chunks
    const int rem   = T - nfull * CHUNK;   // tail rows

    // ---- prologue: async-stage chunk 0 (full, or just the tail if T<CHUNK) ----
    if (valid) {
        const int rows0     = (nfull > 0) ? CHUNK : rem;
        const float*   src  = x + b;
        const uint32_t loff = lds0 + (uint32_t)tid * 4u;
        for (int j = 0; j < rows0; ++j)
            async_ld_f32(src + (size_t)j * B, loff + (uint32_t)j * (BLK * 4u));
    }
    wait_async0();
    __syncthreads();

    // ---- steady state: guard-free full chunks ----
    for (int k = 0; k < nfull; ++k) {
        const int cur = k & 1;

        // issue async stage of the next chunk into the other buffer
        if (valid) {
            const uint32_t lds_nxt = (cur ? lds0 : lds1) + (uint32_t)tid * 4u;
            const float*   src     = x + (size_t)(k + 1) * CHUNK * B + b;
            if (k + 1 < nfull) {
#pragma unroll
                for (int j = 0; j < CHUNK; ++j)
                    async_ld_f32(src + (size_t)j * B, lds_nxt + (uint32_t)j * (BLK * 4u));
            } else {
                for (int j = 0; j < rem; ++j)   // tail stage (cold path)
                    async_ld_f32(src + (size_t)j * B, lds_nxt + (uint32_t)j * (BLK * 4u));
            }
        }

        // consume current full chunk: 32 ds_loads, then pure VALU recurrence
        if (valid) {
            float xv[CHUNK];
            const float* sb = &sbuf[cur][tid];
#pragma unroll
            for (int j = 0; j < CHUNK; ++j)
                xv[j] = sb[j * BLK];

            float* obase = out + (size_t)k * CHUNK * B + b;
#pragma unroll
            for (int j = 0; j < CHUNK; ++j) {
                const float xx = xv[j];
                float zi = fmaf(h, si_h, xx * si_x);
                float zf = fmaf(h, sf_h, xx * sf_x);
                float zo = fmaf(h, so_h, xx * so_x);
                float zg = fmaf(h, g_h,  xx * g_x);
                float ti = fast_tanh(zi);
                float tf = fast_tanh(zf);
                float to = fast_tanh(zo);
                float tg = fast_tanh(zg);
                float sig_i = fmaf(ti, 0.5f, 0.5f);
                float sig_f = fmaf(tf, 0.5f, 0.5f);
                float sig_o = fmaf(to, 0.5f, 0.5f);
                c = fmaf(sig_f, c, sig_i * tg);
                float tc = fast_tanh(c);
                h = sig_o * tc;
                __builtin_nontemporal_store(h, obase + (size_t)j * B);
            }
        }

        // drain this round's async stage, then make it visible block-wide
        wait_async0();
        __syncthreads();
    }

    // ---- tail chunk (rem rows), already staged and synced ----
    if (rem > 0 && valid) {
        const int cur = nfull & 1;
        const float* sb = &sbuf[cur][tid];
        float* obase = out + (size_t)nfull * CHUNK * B + b;
        for (int j = 0; j < rem; ++j) {
            const float xx = sb[j * BLK];
            float zi = fmaf(h, si_h, xx * si_x);
            float zf = fmaf(h, sf_h, xx * sf_x);
            float zo = fmaf(h, so_h, xx * so_x);
            float zg = fmaf(h, g_h,  xx * g_x);
            float ti = fast_tanh(zi);
            float tf = fast_tanh(zf);
            float to = fast_tanh(zo);
            float tg = fast_tanh(zg);
            float sig_i = fmaf(ti, 0.5f, 0.5f);
            float sig_f = fmaf(tf, 0.5f, 0.5f);
            float sig_o = fmaf(to, 0.5f, 0.5f);
            c = fmaf(sig_f, c, sig_i * tg);
            float tc = fast_tanh(c);
            h = sig_o * tc;
            __builtin_nontemporal_store(h, obase + (size_t)j * B);
        }
    }

    if (valid) { hN[b] = h; cN[b] = c; }
}

// ---------------------------------------------------------------------------
extern "C" void kernel_launch(void* const* d_in, const int* in_sizes, int n_in,
                              void* d_out, int out_size, void* d_ws, size_t ws_size,
                              hipStream_t stream) {
    const float* x    = (const float*)d_in[0];
    const float* h0   = (const float*)d_in[1];
    const float* c0   = (const float*)d_in[2];
    const float* w_ih = (const float*)d_in[3];
    const float* w_hh = (const float*)d_in[4];

    const int B = in_sizes[1];            // h0 is [1,B,1]
    const int T = in_sizes[0] / B;        // x is [T,B,1]

    float* out = (float*)d_out;           // [T,B]
    float* hN  = out + (size_t)T * B;     // [B]
    float* cN  = hN + B;                  // [B]

    const int blocks = (B + BLK - 1) / BLK;
    lstm_scan_kernel<<<blocks, BLK, 0, stream>>>(x, h0, c0, w_ih, w_hh,
                                                 out, hN, cN, T, B);
}